// SmoothLoss_9363028705512
// MI455X (gfx1250) — compile-verified
//
#include <hip/hip_runtime.h>
#include <cstdint>

// Problem constants (reference: B=16, C=3, H=W=512, fp32)
#define BATCH 16
#define IMG_H 512
#define IMG_W 512
#define HW    (IMG_H * IMG_W)
#define CHW   (3 * HW)

// Tile geometry: 16x64 interior, halo rows only on top (all offsets have dh>=0),
// halo cols on both sides.
#define TH    16
#define TW    64
#define ROWS  (TH + 2)          // 18: rows -2..15 relative to tile top
#define PITCH (TW + 4)          // 68: cols -2..65 relative to tile left
#define NPIX  (ROWS * PITCH)    // 1224 halo pixels
#define NELEM (3 * NPIX)        // 3672 scalar loads per array
#define NBLOCKS (BATCH * (IMG_H / TH) * (IMG_W / TW))  // 4096
#define NCHUNK ((NPIX + 15) / 16)                      // 77 WMMA chunks

typedef __attribute__((ext_vector_type(2))) float v2f;
typedef __attribute__((ext_vector_type(8))) float v8f;

// ---- CDNA5 async global->LDS DMA (gfx1250) --------------------------------
__device__ __forceinline__ void async_b32(const float* gptr, uint32_t lds_byte_addr) {
    asm volatile("global_load_async_to_lds_b32 %0, %1, off"
                 :: "v"(lds_byte_addr), "v"(gptr)
                 : "memory");
}
__device__ __forceinline__ void wait_async0() {
    asm volatile("s_wait_asynccnt 0" ::: "memory");
}
// ---------------------------------------------------------------------------

__global__ __launch_bounds__(256)
void smooth_tile_kernel(const float* __restrict__ in_rgb,
                        const float* __restrict__ out_img,
                        float* __restrict__ partial) {
    // Channel-interleaved (x4 for 16B alignment) halo tiles.
    __shared__ __align__(16) float xs[NPIX * 4];
    __shared__ __align__(16) float ys[NPIX * 4];
    __shared__ float btab[64];   // WMMA B-operand lane table (2 VGPRs x 32 lanes)
    __shared__ float red[8];

    const int tid = threadIdx.x;
    const int th0 = blockIdx.y * TH;
    const int tw0 = blockIdx.x * TW;

    const float* xin = in_rgb  + (size_t)blockIdx.z * CHW;
    const float* yin = out_img + (size_t)blockIdx.z * CHW;

    const uint32_t xs_base = (uint32_t)(uintptr_t)(void*)xs;
    const uint32_t ys_base = (uint32_t)(uintptr_t)(void*)ys;

    // ---- Stage both tiles (with halo) into LDS via async DMA ----
    for (int t = tid; t < NELEM; t += 256) {
        const int c  = t / NPIX;
        const int rj = t - c * NPIX;
        const int r  = rj / PITCH;
        const int j  = rj - r * PITCH;
        int gh = th0 - 2 + r; gh = gh < 0 ? 0 : gh;             // top clamp only
        int gw = tw0 - 2 + j; gw = gw < 0 ? 0 : (gw > IMG_W - 1 ? IMG_W - 1 : gw);
        const size_t   goff = (size_t)c * HW + (size_t)gh * IMG_W + (size_t)gw;
        const uint32_t loff = (uint32_t)((rj * 4 + c) << 2);
        async_b32(xin + goff, xs_base + loff);
        async_b32(yin + goff, ys_base + loff);
    }

    // B-operand table for V_WMMA_F32_16X16X4_F32 (B is 4x16; VGPR0 holds rows
    // K0 | K2 across lane halves, VGPR1 holds rows K1 | K3(bias)).
    // D[m,k] = R*mat[0][k] + G*mat[1][k] + B*mat[2][k] + bias[k]
    if (tid < 64) {
        constexpr float tab[64] = {
            // b0, lanes 0-15: mat row 0 (K=0) over output cols n
            0.257f, -0.148f, 0.439f, 0.f, 0.f, 0.f, 0.f, 0.f,
            0.f, 0.f, 0.f, 0.f, 0.f, 0.f, 0.f, 0.f,
            // b0, lanes 16-31: mat row 2 (K=2)
            0.098f, 0.439f, -0.071f, 0.f, 0.f, 0.f, 0.f, 0.f,
            0.f, 0.f, 0.f, 0.f, 0.f, 0.f, 0.f, 0.f,
            // b1, lanes 0-15: mat row 1 (K=1)
            0.564f, -0.291f, -0.368f, 0.f, 0.f, 0.f, 0.f, 0.f,
            0.f, 0.f, 0.f, 0.f, 0.f, 0.f, 0.f, 0.f,
            // b1, lanes 16-31: bias row (K=3, multiplies A's 1.0)
            16.0f / 255.0f, 128.0f / 255.0f, 128.0f / 255.0f, 0.f, 0.f, 0.f, 0.f, 0.f,
            0.f, 0.f, 0.f, 0.f, 0.f, 0.f, 0.f, 0.f
        };
        btab[tid] = tab[tid];
    }

    wait_async0();
    __syncthreads();

    // ---- RGB -> YCbCr in place via V_WMMA_F32_16X16X4_F32 -----------------
    // Each wave converts 16 halo pixels per WMMA: A(16x4) = [R,G,B,1] rows,
    // B(4x16) = conversion matrix + bias, D[m,0..2] = {Y,Cb,Cr}.
    {
        const int wave = tid >> 5;
        const int lane = tid & 31;
        const int sub  = lane & 15;
        const int hi   = lane >> 4;

        v2f bmat;
        bmat.x = btab[lane];
        bmat.y = btab[lane + 32];

        for (int ch = wave; ch < NCHUNK; ch += 8) {       // wave-uniform bound
            const int p0 = ch * 16;
            int P = p0 + sub; P = P > NPIX - 1 ? NPIX - 1 : P;
            // lo lanes: {R,G} at pixel base; hi lanes: {B, pad} at base+8B
            const float2 rg = *reinterpret_cast<const float2*>(&xs[P * 4 + hi * 2]);
            v2f a;
            a.x = rg.x;
            a.y = hi ? 1.0f : rg.y;
            v8f c = {};
            v8f d = __builtin_amdgcn_wmma_f32_16x16x4_f32(
                false, a, false, bmat, (short)0, c, false, false);
            // Scatter D[m, n<3] back: VGPR r holds M=r (lanes 0-15) / M=r+8.
            if (sub < 3) {
                #pragma unroll
                for (int r2 = 0; r2 < 8; ++r2) {
                    int P2 = p0 + r2 + hi * 8;
                    P2 = P2 > NPIX - 1 ? NPIX - 1 : P2;
                    xs[P2 * 4 + sub] = d[r2];
                }
            }
        }
    }
    __syncthreads();

    // ---- Stencil: 12 unique offsets (± pairs have equal means -> 2x weight)
    constexpr float sExp = -1.0f / (2.0f * 10.0f * 10.0f);
    constexpr int DHS[12] = {0, 0, 1, 1, 1, 1, 1, 2, 2, 2, 2, 2};
    constexpr int DWS[12] = {1, 2, -2, -1, 0, 1, 2, -2, -1, 0, 1, 2};

    float acc = 0.0f;

    #pragma unroll
    for (int k = 0; k < (TH * TW) / 256; ++k) {
        const int pix = tid + k * 256;
        const int li  = pix >> 6;        // row in tile (TW == 64)
        const int lj  = pix & 63;
        const int r   = li + 2;          // halo row
        const int jc  = lj + 2;          // halo col
        const int gh  = th0 + li;
        const int gw  = tw0 + lj;

        const float4 xc = *reinterpret_cast<const float4*>(&xs[(r * PITCH + jc) * 4]);
        const float4 yc = *reinterpret_cast<const float4*>(&ys[(r * PITCH + jc) * 4]);

        #pragma unroll
        for (int oi = 0; oi < 12; ++oi) {
            const int dh  = DHS[oi];
            const int dw  = DWS[oi];
            const int adw = dw < 0 ? -dw : dw;
            // 2x (pair symmetry) / (24 * B * (H-|dh|) * (W-|dw|)) -- compile-time
            const float wgt = 1.0f / (12.0f * (float)BATCH *
                                      (float)(IMG_H - dh) * (float)(IMG_W - adw));
            const int qh = gh - dh;
            const int qw = gw - dw;
            const int nidx = ((r - dh) * PITCH + (jc - dw)) * 4;

            const float4 xn = *reinterpret_cast<const float4*>(&xs[nidx]);
            const float4 yn = *reinterpret_cast<const float4*>(&ys[nidx]);

            const float d0 = xc.x - xn.x;
            const float d1 = xc.y - xn.y;
            const float d2 = xc.z - xn.z;
            const float dist = fmaf(d0, d0, fmaf(d1, d1, d2 * d2));
            const float wv = __expf(dist * sExp);
            const float l1 = fabsf(yc.x - yn.x) + fabsf(yc.y - yn.y) + fabsf(yc.z - yn.z);

            const bool ok = (qh >= 0) & (qw >= 0) & (qw < IMG_W);
            acc += ok ? (wgt * wv * l1) : 0.0f;
        }
    }

    // ---- Deterministic block reduction: wave32 shuffle tree, then LDS ----
    #pragma unroll
    for (int off = 16; off > 0; off >>= 1)
        acc += __shfl_down(acc, off, 32);
    if ((tid & 31) == 0) red[tid >> 5] = acc;
    __syncthreads();
    if (tid == 0) {
        float s = 0.0f;
        #pragma unroll
        for (int i = 0; i < 8; ++i) s += red[i];
        const int bid = (blockIdx.z * gridDim.y + blockIdx.y) * gridDim.x + blockIdx.x;
        partial[bid] = s;
    }
}

// Single-block, fixed-order final reduction (bitwise deterministic).
__global__ __launch_bounds__(256)
void smooth_reduce_kernel(const float* __restrict__ partial, float* __restrict__ out) {
    __shared__ float red[256];
    const int t = threadIdx.x;
    float s = 0.0f;
    for (int i = t; i < NBLOCKS; i += 256) s += partial[i];
    red[t] = s;
    __syncthreads();
    #pragma unroll
    for (int stride = 128; stride > 0; stride >>= 1) {
        if (t < stride) red[t] += red[t + stride];
        __syncthreads();
    }
    if (t == 0) out[0] = red[0];
}

extern "C" void kernel_launch(void* const* d_in, const int* in_sizes, int n_in,
                              void* d_out, int out_size, void* d_ws, size_t ws_size,
                              hipStream_t stream) {
    const float* input  = (const float*)d_in[0];
    const float* output = (const float*)d_in[1];
    float* ws  = (float*)d_ws;
    float* out = (float*)d_out;

    dim3 grid(IMG_W / TW, IMG_H / TH, BATCH);   // 8 x 32 x 16 = 4096 blocks
    smooth_tile_kernel<<<grid, 256, 0, stream>>>(input, output, ws);
    smooth_reduce_kernel<<<1, 256, 0, stream>>>(ws, out);
}